// EquivariantGraphConv_35837207117863
// MI455X (gfx1250) — compile-verified
//
#include <hip/hip_runtime.h>

#define N_NODES  100000
#define N_EDGES  1600000
#define IN_CH    128
#define OUT_CH   64
#define EDGE_DIM 32

typedef float v2f __attribute__((ext_vector_type(2)));
typedef float v8f __attribute__((ext_vector_type(8)));

__device__ __forceinline__ void f32_atomic_add(float* p, float v) {
    __hip_atomic_fetch_add(p, v, __ATOMIC_RELAXED, __HIP_MEMORY_SCOPE_AGENT);
}

__device__ __forceinline__ v8f wmma_f32(v2f a, v2f b, v8f c) {
    return __builtin_amdgcn_wmma_f32_16x16x4_f32(
        false, a, false, b, (short)0, c, false, false);
}

// ---------------------------------------------------------------------------
// Zero-fill (deterministic init of d_out / cnt each call)
// ---------------------------------------------------------------------------
__global__ void zero_kernel(float* __restrict__ a, int n) {
    for (int i = blockIdx.x * blockDim.x + threadIdx.x; i < n;
         i += gridDim.x * blockDim.x)
        a[i] = 0.0f;
}

// ---------------------------------------------------------------------------
// h = x @ W_node + b_node   [N_NODES,128] x [128,64] -> [N_NODES,64]
// One wave per 32-row tile (2 M-tiles share every B fragment), 4 N-tiles,
// K via V_WMMA_F32_16X16X4_F32 (fp32 exact wrt reference).
// ---------------------------------------------------------------------------
__global__ void node_gemm_kernel(const float* __restrict__ x,
                                 const float* __restrict__ Wn,
                                 const float* __restrict__ bn,
                                 float* __restrict__ h) {
    const int lane = threadIdx.x & 31;
    const int wib  = threadIdx.x >> 5;
    const int wpb  = blockDim.x >> 5;
    const int lh   = lane >> 4;      // half-wave: K sub-pair select / M+8
    const int nlo  = lane & 15;      // M index (A), N index (B/C)
    const int nTiles = N_NODES / 32; // 3125, exact

    float bias[4];
#pragma unroll
    for (int nt = 0; nt < 4; ++nt) bias[nt] = bn[nt * 16 + nlo];

    for (int tile = blockIdx.x * wpb + wib; tile < nTiles;
         tile += gridDim.x * wpb) {
        const int m0 = tile * 32;

        v8f acc[2][4];
#pragma unroll
        for (int mt = 0; mt < 2; ++mt)
#pragma unroll
            for (int nt = 0; nt < 4; ++nt)
#pragma unroll
                for (int r = 0; r < 8; ++r) acc[mt][nt][r] = bias[nt];

        const float* a0p = x + (size_t)(m0 + nlo) * IN_CH;
        const float* a1p = a0p + 16 * IN_CH;

#pragma unroll 4
        for (int ks = 0; ks < IN_CH / 4; ++ks) {
            const int k0 = ks * 4 + lh * 2;
            // A 16x4 f32: lanes 0-15 hold K=k0,k0+1; lanes 16-31 K=k0+2,k0+3
            const v2f a0 = *(const v2f*)(a0p + k0);
            const v2f a1 = *(const v2f*)(a1p + k0);
#pragma unroll
            for (int nt = 0; nt < 4; ++nt) {
                const float* bp = Wn + k0 * OUT_CH + nt * 16 + nlo;
                v2f b; b.x = bp[0]; b.y = bp[OUT_CH];
                acc[0][nt] = wmma_f32(a0, b, acc[0][nt]);
                acc[1][nt] = wmma_f32(a1, b, acc[1][nt]);
            }
        }

        // C/D layout: VGPR r -> M=r (lanes 0-15) / M=r+8 (lanes 16-31), N=nlo
#pragma unroll
        for (int mt = 0; mt < 2; ++mt)
#pragma unroll
            for (int nt = 0; nt < 4; ++nt)
#pragma unroll
                for (int r = 0; r < 8; ++r) {
                    const int off = (m0 + mt * 16 + r + 8 * lh) * OUT_CH +
                                    nt * 16 + nlo;   // < 6.4M, 32-bit safe
                    h[off] = acc[mt][nt][r];
                }
    }
}

// ---------------------------------------------------------------------------
// e = edge_attr @ W_edge + b_edge; msg = h[col] + e; atomic scatter-sum into
// out[row]; per-edge count. Persistent waves: W_edge (8 KB) is preloaded as
// 8x4 register-resident B fragments, reused across all edge tiles of a wave.
// ---------------------------------------------------------------------------
__global__ void edge_gemm_scatter_kernel(const float* __restrict__ ea,
                                         const float* __restrict__ We,
                                         const float* __restrict__ be,
                                         const int*   __restrict__ ei,
                                         const float* __restrict__ h,
                                         float* __restrict__ outsum,
                                         float* __restrict__ cnt) {
    const int lane = threadIdx.x & 31;
    const int wib  = threadIdx.x >> 5;
    const int wpb  = blockDim.x >> 5;
    const int lh   = lane >> 4;
    const int nlo  = lane & 15;
    const int nTiles = N_EDGES / 16; // 100000, exact

    // Hoist entire W_edge into registers: 8 k-steps x 4 N-tiles x v2f.
    v2f bfrag[EDGE_DIM / 4][4];
    float bias[4];
#pragma unroll
    for (int ks = 0; ks < EDGE_DIM / 4; ++ks) {
        const int k0 = ks * 4 + lh * 2;
#pragma unroll
        for (int nt = 0; nt < 4; ++nt) {
            const float* bp = We + k0 * OUT_CH + nt * 16 + nlo;
            bfrag[ks][nt].x = bp[0];
            bfrag[ks][nt].y = bp[OUT_CH];
        }
    }
#pragma unroll
    for (int nt = 0; nt < 4; ++nt) bias[nt] = be[nt * 16 + nlo];

    for (int tile = blockIdx.x * wpb + wib; tile < nTiles;
         tile += gridDim.x * wpb) {
        const int e0 = tile * 16;

        v8f acc[4];
#pragma unroll
        for (int nt = 0; nt < 4; ++nt)
#pragma unroll
            for (int r = 0; r < 8; ++r) acc[nt][r] = bias[nt];

        const float* ap = ea + (size_t)(e0 + nlo) * EDGE_DIM;
#pragma unroll
        for (int ks = 0; ks < EDGE_DIM / 4; ++ks) {
            const v2f a = *(const v2f*)(ap + ks * 4 + lh * 2);
#pragma unroll
            for (int nt = 0; nt < 4; ++nt)
                acc[nt] = wmma_f32(a, bfrag[ks][nt], acc[nt]);
        }

        // edge_index is [2, E] flat: row = ei[e], col = ei[E + e]
        int ob[8], hb[8], rw[8];
#pragma unroll
        for (int r = 0; r < 8; ++r) {
            const int e = e0 + r + 8 * lh;
            rw[r] = ei[e];
            ob[r] = rw[r] * OUT_CH;          // 32-bit offsets (< 6.4M)
            hb[r] = ei[N_EDGES + e] * OUT_CH;
        }

#pragma unroll
        for (int r = 0; r < 8; ++r) {
            if (nlo == 0)   // lanes 0 & 16 together cover all 16 edges once
                f32_atomic_add(&cnt[rw[r]], 1.0f);
            const int c = nlo;
#pragma unroll
            for (int nt = 0; nt < 4; ++nt) {
                const float v = acc[nt][r] + h[hb[r] + nt * 16 + c];
                f32_atomic_add(&outsum[ob[r] + nt * 16 + c], v);
            }
        }
    }
}

// ---------------------------------------------------------------------------
// out = sum / max(cnt, 1)
// ---------------------------------------------------------------------------
__global__ void finalize_kernel(float* __restrict__ out,
                                const float* __restrict__ cnt) {
    const int total = N_NODES * OUT_CH;
    for (int i = blockIdx.x * blockDim.x + threadIdx.x; i < total;
         i += gridDim.x * blockDim.x) {
        const float c = cnt[i >> 6]; // /OUT_CH
        out[i] = out[i] / fmaxf(c, 1.0f);
    }
}

extern "C" void kernel_launch(void* const* d_in, const int* in_sizes, int n_in,
                              void* d_out, int out_size, void* d_ws, size_t ws_size,
                              hipStream_t stream) {
    const float* x  = (const float*)d_in[0];
    const int*   ei = (const int*)  d_in[1];
    const float* ea = (const float*)d_in[2];
    const float* Wn = (const float*)d_in[3];
    const float* bn = (const float*)d_in[4];
    const float* We = (const float*)d_in[5];
    const float* be = (const float*)d_in[6];
    float* out = (float*)d_out;

    // Workspace: h [N_NODES*64] floats, then cnt [N_NODES] floats (~26 MB)
    float* h   = (float*)d_ws;
    float* cnt = h + (size_t)N_NODES * OUT_CH;

    zero_kernel<<<2048, 256, 0, stream>>>(out, N_NODES * OUT_CH);
    zero_kernel<<<256, 256, 0, stream>>>(cnt, N_NODES);

    // node GEMM: 3125 wave-tiles (32 rows each), 8 waves/block
    node_gemm_kernel<<<(N_NODES / 32 + 7) / 8, 256, 0, stream>>>(x, Wn, bn, h);

    // edge GEMM + fused gather/scatter: persistent waves, ~5 tiles/wave so
    // the register-resident W_edge fragments amortize.
    edge_gemm_scatter_kernel<<<2500, 256, 0, stream>>>(
        ea, We, be, ei, h, out, cnt);

    finalize_kernel<<<2048, 256, 0, stream>>>(out, cnt);
}